// ALIGNNMT_5823975653421
// MI455X (gfx1250) — compile-verified
//
#include <hip/hip_runtime.h>
#include <math.h>

// GatedGCN layer for MI455X (gfx1250, wave32).
// fp32 WMMA (V_WMMA_F32_16X16X4_F32) for all GEMMs; fused gather/scatter epilogues.
// LDS staging via GLOBAL_LOAD_ASYNC_TO_LDS_B128 (ASYNCcnt path) when available,
// else vectorized b128 copies.

typedef float v2f __attribute__((ext_vector_type(2)));
typedef float v8f __attribute__((ext_vector_type(8)));

#define DD 64

#if defined(__has_builtin)
#if __has_builtin(__builtin_amdgcn_global_load_async_to_lds_b128)
#define HAVE_ASYNC_LDS 1
#endif
#endif
#ifndef HAVE_ASYNC_LDS
#define HAVE_ASYNC_LDS 0
#endif

__device__ __forceinline__ float sigmoidf_(float x) {
    return 1.0f / (1.0f + __expf(-x));
}

#if HAVE_ASYNC_LDS
// Builtin signature (from clang diagnostic): pointers to 16B int vectors,
// global (AS1) source, LDS (AS3) destination, imm offset, imm cpol.
typedef int av4i __attribute__((vector_size(16)));
typedef __attribute__((address_space(1))) av4i* gptr_t;
typedef __attribute__((address_space(3))) av4i* lptr_t;

__device__ __forceinline__ void async_copy16(const float* g, float* l) {
    __builtin_amdgcn_global_load_async_to_lds_b128(
        (gptr_t)(g), (lptr_t)(l), 0, 0);
}
__device__ __forceinline__ void wait_async0() {
#if __has_builtin(__builtin_amdgcn_s_wait_asynccnt)
    __builtin_amdgcn_s_wait_asynccnt(0);
#else
    asm volatile("s_wait_asynccnt 0x0" ::: "memory");
#endif
}
#else
__device__ __forceinline__ void async_copy16(const float* g, float* l) {
    *(float4*)l = *(const float4*)g;
}
__device__ __forceinline__ void wait_async0() {}
#endif

// ---------------------------------------------------------------------------
// Kernel 1: four fused node-side GEMMs: [N,64] @ [64,64] + bias
//   wave0 -> e_src = nf@Wsg+bsg   wave1 -> e_dst = nf@Wdg+bdg
//   wave2 -> Bh    = nf@Wdu+bdu   wave3 -> Cx    = nf@Wsu+bsu
// One block (4 waves) per 16-row tile; A tile shared via LDS.
// ---------------------------------------------------------------------------
__global__ void __launch_bounds__(128)
node_gemm4(const float* __restrict__ nf, int N,
           const float* __restrict__ Wsg, const float* __restrict__ bsg,
           const float* __restrict__ Wdg, const float* __restrict__ bdg,
           const float* __restrict__ Wdu, const float* __restrict__ bdu,
           const float* __restrict__ Wsu, const float* __restrict__ bsu,
           float* __restrict__ e_src, float* __restrict__ e_dst,
           float* __restrict__ Bh,    float* __restrict__ Cx)
{
    __shared__ __align__(16) float As[16][68];  // 68-stride: bank-conflict free

    const int tid  = threadIdx.x;
    const int row0 = blockIdx.x * 16;
    const bool full = (row0 + 16 <= N);         // block-uniform

    if (full) {
        #pragma unroll
        for (int t = 0; t < 2; ++t) {           // 256 float4s / 128 threads
            const int idx = tid + t * 128;
            const int r = idx >> 4, c4 = (idx & 15) * 4;
            async_copy16(&nf[(size_t)(row0 + r) * DD + c4], &As[r][c4]);
        }
        wait_async0();
    } else {
        for (int idx = tid; idx < 16 * DD; idx += 128) {
            const int r = idx >> 6, c = idx & 63;
            As[r][c] = (row0 + r < N) ? nf[(size_t)(row0 + r) * DD + c] : 0.0f;
        }
    }
    __syncthreads();

    const int wave = tid >> 5;
    const int lane = tid & 31;
    const int lh   = lane >> 4;   // lane half: 0 or 1
    const int j    = lane & 15;

    const float* W  = (wave == 0) ? Wsg : (wave == 1) ? Wdg : (wave == 2) ? Wdu : Wsu;
    const float* bi = (wave == 0) ? bsg : (wave == 1) ? bdg : (wave == 2) ? bdu : bsu;
    float*     outp = (wave == 0) ? e_src : (wave == 1) ? e_dst : (wave == 2) ? Bh : Cx;

    for (int n = 0; n < 4; ++n) {
        const int cb = n * 16 + j;
        const float bval = bi[cb];
        v8f acc;
        #pragma unroll
        for (int r = 0; r < 8; ++r) acc[r] = bval;

        #pragma unroll
        for (int k = 0; k < DD; k += 4) {
            const int k0 = k + 2 * lh;
            v2f a, b;
            // A 16x4 layout: lane j (rows 0-15), VGPR0/1 = K {2*lh, 2*lh+1}
            a.x = As[j][k0];
            a.y = As[j][k0 + 1];
            // B 4x16 layout: lane j -> col, VGPR0/1 = K {2*lh, 2*lh+1}
            b.x = W[(size_t)k0 * DD + cb];
            b.y = W[(size_t)(k0 + 1) * DD + cb];
            acc = __builtin_amdgcn_wmma_f32_16x16x4_f32(
                false, a, false, b, (short)0, acc, false, false);
        }

        if (full) {
            #pragma unroll
            for (int r = 0; r < 8; ++r)
                outp[(size_t)(row0 + r + 8 * lh) * DD + cb] = acc[r];
        } else {
            #pragma unroll
            for (int r = 0; r < 8; ++r) {
                const int row = row0 + r + 8 * lh;   // C/D layout: M = r + 8*half
                if (row < N) outp[(size_t)row * DD + cb] = acc[r];
            }
        }
    }
}

// ---------------------------------------------------------------------------
// Kernel 2: per-edge gate. One wave per 16-edge tile, 8 waves/block.
//   m = (ef@Weg+beg) + e_src[src] + e_dst[dst]  (WMMA GEMM + gather)
//   sigma = sigmoid(m); atomics into sum_sigma_h / sum_sigma on dst;
//   BN stats of m via LDS reduction -> global atomics.
//   m is written to m_out (the y-half of d_out) for the later BN pass.
// ---------------------------------------------------------------------------
__global__ void __launch_bounds__(256)
edge_gate(const float* __restrict__ ef, int E,
          const int* __restrict__ src, const int* __restrict__ dst,
          const float* __restrict__ Weg, const float* __restrict__ beg,
          const float* __restrict__ e_src, const float* __restrict__ e_dst,
          const float* __restrict__ Bh,
          float* __restrict__ m_out, float* __restrict__ ssh,
          float* __restrict__ ssum, float* __restrict__ estats)
{
    __shared__ __align__(16) float Ws[64][68];     // Weg staged for all 8 waves
    __shared__ __align__(16) float Es[8][16][68];  // per-wave edge_feats tile
    __shared__ int   sS[8][16], sD[8][16];
    __shared__ float bn[128];          // [0:64) sum(m), [64:128) sum(m^2)

    const int tid = threadIdx.x;
    if (tid < 128) bn[tid] = 0.0f;
    #pragma unroll
    for (int t = 0; t < 4; ++t) {      // 1024 float4s / 256 threads
        const int idx = tid + t * 256;
        const int r = idx >> 4, c4 = (idx & 15) * 4;
        async_copy16(&Weg[(size_t)r * DD + c4], &Ws[r][c4]);
    }

    const int wave = tid >> 5;
    const int lane = tid & 31;
    const int lh   = lane >> 4;
    const int j    = lane & 15;
    const int e0   = (blockIdx.x * 8 + wave) * 16;
    const bool fullTile = (e0 + 16 <= E);          // wave-uniform

    if (fullTile) {
        #pragma unroll
        for (int t = 0; t < 8; ++t) {  // 256 float4s / 32 lanes
            const int idx = lane + t * 32;
            const int r = idx >> 4, c4 = (idx & 15) * 4;
            async_copy16(&ef[(size_t)(e0 + r) * DD + c4], &Es[wave][r][c4]);
        }
    } else {
        for (int idx = lane; idx < 16 * DD; idx += 32) {
            const int r = idx >> 6, c = idx & 63;
            Es[wave][r][c] = (e0 + r < E) ? ef[(size_t)(e0 + r) * DD + c] : 0.0f;
        }
    }
    if (lane < 16) {
        const int e = e0 + lane;
        sS[wave][lane] = (e < E) ? src[e] : 0;
        sD[wave][lane] = (e < E) ? dst[e] : 0;
    }
    wait_async0();
    __syncthreads();

    float sumv[4] = {0.f, 0.f, 0.f, 0.f};
    float sqv[4]  = {0.f, 0.f, 0.f, 0.f};

    #pragma unroll
    for (int n = 0; n < 4; ++n) {
        const int cb = n * 16 + j;
        const float bval = beg[cb];
        v8f acc;
        #pragma unroll
        for (int r = 0; r < 8; ++r) acc[r] = bval;

        #pragma unroll
        for (int k = 0; k < DD; k += 4) {
            const int k0 = k + 2 * lh;
            v2f a, b;
            a.x = Es[wave][j][k0];
            a.y = Es[wave][j][k0 + 1];
            b.x = Ws[k0][cb];
            b.y = Ws[k0 + 1][cb];
            acc = __builtin_amdgcn_wmma_f32_16x16x4_f32(
                false, a, false, b, (short)0, acc, false, false);
        }

        if (fullTile) {
            #pragma unroll
            for (int r = 0; r < 8; ++r) {
                const int rr = r + 8 * lh;
                const int ss = sS[wave][rr];
                const int sd = sD[wave][rr];
                float mval = acc[r] + e_src[(size_t)ss * DD + cb]
                                    + e_dst[(size_t)sd * DD + cb];
                m_out[(size_t)(e0 + rr) * DD + cb] = mval;
                const float sg = sigmoidf_(mval);
                sumv[n] += mval;
                sqv[n]  += mval * mval;
                atomicAdd(&ssum[(size_t)sd * DD + cb], sg);
                atomicAdd(&ssh [(size_t)sd * DD + cb], sg * Bh[(size_t)ss * DD + cb]);
            }
        } else {
            #pragma unroll
            for (int r = 0; r < 8; ++r) {
                const int rr = r + 8 * lh;
                const int e  = e0 + rr;
                if (e < E) {
                    const int ss = sS[wave][rr];
                    const int sd = sD[wave][rr];
                    float mval = acc[r] + e_src[(size_t)ss * DD + cb]
                                        + e_dst[(size_t)sd * DD + cb];
                    m_out[(size_t)e * DD + cb] = mval;
                    const float sg = sigmoidf_(mval);
                    sumv[n] += mval;
                    sqv[n]  += mval * mval;
                    atomicAdd(&ssum[(size_t)sd * DD + cb], sg);
                    atomicAdd(&ssh [(size_t)sd * DD + cb], sg * Bh[(size_t)ss * DD + cb]);
                }
            }
        }
    }

    #pragma unroll
    for (int n = 0; n < 4; ++n) {
        atomicAdd(&bn[n * 16 + j],      sumv[n]);
        atomicAdd(&bn[64 + n * 16 + j], sqv[n]);
    }
    __syncthreads();
    if (tid < 128) atomicAdd(&estats[tid], bn[tid]);
}

// ---------------------------------------------------------------------------
// Kernel 3: h = ssh/(ssum+1e-6); x_pre = Cx + h (into x-half of d_out);
//           accumulate BN stats for x via LDS -> global atomics.
// ---------------------------------------------------------------------------
__global__ void __launch_bounds__(256)
node_combine(const float* __restrict__ Cx, const float* __restrict__ ssh,
             const float* __restrict__ ssum, float* __restrict__ xpre,
             float* __restrict__ nstats, int total)
{
    __shared__ float bn[128];
    const int tid = threadIdx.x;
    if (tid < 128) bn[tid] = 0.0f;
    __syncthreads();

    const int idx = blockIdx.x * 256 + tid;
    if (idx < total) {
        const int col = idx & 63;
        const float h  = ssh[idx] / (ssum[idx] + 1e-6f);
        const float xp = Cx[idx] + h;
        xpre[idx] = xp;
        atomicAdd(&bn[col],      xp);
        atomicAdd(&bn[64 + col], xp * xp);
    }
    __syncthreads();
    if (tid < 128) atomicAdd(&nstats[tid], bn[tid]);
}

// ---------------------------------------------------------------------------
// Kernels 4/5: out = base + silu( BN(pre) ), in place on `pre_inout`.
// mean = S/n ; var = Sq/n - mean^2 (biased, matches reference).
// ---------------------------------------------------------------------------
__global__ void __launch_bounds__(256)
bn_silu_residual(const float* __restrict__ base, float* __restrict__ pre_inout,
                 const float* __restrict__ stats, const float* __restrict__ gamma,
                 const float* __restrict__ beta, float invCount, int total)
{
    const int idx = blockIdx.x * 256 + threadIdx.x;
    if (idx >= total) return;
    const int col = idx & 63;
    const float mean = stats[col] * invCount;
    const float var  = stats[64 + col] * invCount - mean * mean;
    const float xn = (pre_inout[idx] - mean) * rsqrtf(var + 1e-5f) * gamma[col]
                   + beta[col];
    pre_inout[idx] = base[idx] + xn * sigmoidf_(xn);
}

// ---------------------------------------------------------------------------
extern "C" void kernel_launch(void* const* d_in, const int* in_sizes, int n_in,
                              void* d_out, int out_size, void* d_ws, size_t ws_size,
                              hipStream_t stream)
{
    const float* nf  = (const float*)d_in[0];
    const float* ef  = (const float*)d_in[1];
    const int*   src = (const int*)  d_in[2];
    const int*   dst = (const int*)  d_in[3];
    const float* Wsg = (const float*)d_in[4];  const float* bsg = (const float*)d_in[5];
    const float* Wdg = (const float*)d_in[6];  const float* bdg = (const float*)d_in[7];
    const float* Weg = (const float*)d_in[8];  const float* beg = (const float*)d_in[9];
    const float* Wsu = (const float*)d_in[10]; const float* bsu = (const float*)d_in[11];
    const float* Wdu = (const float*)d_in[12]; const float* bdu = (const float*)d_in[13];
    const float* gn  = (const float*)d_in[14]; const float* btn = (const float*)d_in[15];
    const float* ge  = (const float*)d_in[16]; const float* bte = (const float*)d_in[17];

    const int N = in_sizes[0] / DD;
    const int E = in_sizes[1] / DD;
    const size_t ND = (size_t)N * DD;
    const size_t ED = (size_t)E * DD;

    float* out_x = (float*)d_out;        // [N,64] final x (holds x_pre first)
    float* out_y = out_x + ND;           // [E,64] final y (holds m first)

    // workspace layout (floats): ssh[ND] | ssum[ND] | estats[128] | nstats[128]
    //                            | e_src[ND] | e_dst[ND] | Bh[ND] | Cx[ND]
    float* ws     = (float*)d_ws;
    float* ssh    = ws;
    float* ssum   = ssh + ND;
    float* estats = ssum + ND;
    float* nstats = estats + 128;
    float* e_srcB = nstats + 128;
    float* e_dstB = e_srcB + ND;
    float* BhB    = e_dstB + ND;
    float* CxB    = BhB + ND;

    // zero the accumulators (graph-capturable memset node)
    (void)hipMemsetAsync(ws, 0, (2 * ND + 256) * sizeof(float), stream);

    node_gemm4<<<(N + 15) / 16, 128, 0, stream>>>(
        nf, N, Wsg, bsg, Wdg, bdg, Wdu, bdu, Wsu, bsu,
        e_srcB, e_dstB, BhB, CxB);

    const int etiles = (E + 15) / 16;
    edge_gate<<<(etiles + 7) / 8, 256, 0, stream>>>(
        ef, E, src, dst, Weg, beg, e_srcB, e_dstB, BhB,
        out_y, ssh, ssum, estats);

    node_combine<<<(int)((ND + 255) / 256), 256, 0, stream>>>(
        CxB, ssh, ssum, out_x, nstats, (int)ND);

    bn_silu_residual<<<(int)((ND + 255) / 256), 256, 0, stream>>>(
        nf, out_x, nstats, gn, btn, 1.0f / (float)N, (int)ND);

    bn_silu_residual<<<(int)((ED + 255) / 256), 256, 0, stream>>>(
        ef, out_y, estats, ge, bte, 1.0f / (float)E, (int)ED);
}